// DeformConv2d_85710367359347
// MI455X (gfx1250) — compile-verified
//
#include <hip/hip_runtime.h>

typedef __attribute__((ext_vector_type(2))) float v2f;
typedef __attribute__((ext_vector_type(8))) float v8f;

#define Bn  4
#define Cn  64
#define Hn  128
#define Wn  128
#define On  64
#define K2n 9
#define HoN 128
#define WoN 128
#define CK  (Cn*K2n)     // 576 reduction length
#define HW  (Hn*Wn)
#define PSTR 160         // sampled-tile pair-row stride (words): ≡32 mod 64 banks

// ---------------------------------------------------------------------------
// Repack deform_w (O,C,3,3) into WMMA f32 16x16x4 A-fragment order:
//   id = tap*4096 + cc*256 + ochunk*64 + lane*2 + v
//   value = w[o = ochunk*16 + (lane&15)][c = cc*4 + v + 2*(lane>>4)][tap]
// Per-lane the two K values are contiguous -> one coalesced b64 load later.
// ---------------------------------------------------------------------------
__global__ __launch_bounds__(256) void repack_w(const float* __restrict__ dw,
                                                float* __restrict__ pw) {
  int id = blockIdx.x * 256 + threadIdx.x;
  if (id >= K2n * 16 * 4 * 32 * 2) return;
  int v    = id & 1;
  int lane = (id >> 1) & 31;
  int oc   = (id >> 6) & 3;
  int cc   = (id >> 8) & 15;
  int tap  = id >> 12;
  int o = oc * 16 + (lane & 15);
  int c = cc * 4 + v + 2 * (lane >> 4);
  pw[id] = dw[o * CK + c * K2n + tap];
}

// ---------------------------------------------------------------------------
// Fused: offset conv -> bilinear sampling -> WMMA contraction -> bias+store.
// Block: 128 threads (4 waves). Tile: (batch b, row y, 64-pixel strip).
// Wave w owns output channels [16w, 16w+16) x all 64 pixels (4 accumulators).
// ---------------------------------------------------------------------------
__global__ __launch_bounds__(128) void deform_fused(
    const float* __restrict__ x,  const float* __restrict__ ow,
    const float* __restrict__ ob, const float* __restrict__ pw,
    const float* __restrict__ db, float* __restrict__ out) {
  __shared__ float s_off[18 * 64];                  // offset conv result
  __shared__ int   s_ad[K2n * 64 * 4];              // 4 clamped gather offsets
  __shared__ float s_bw[K2n * 64 * 4];              // validity-folded weights
  __shared__ __align__(16) float sS[32 * PSTR];     // sampled tile, c-paired

  const int bid   = blockIdx.x;
  const int strip = bid & 1;
  const int y     = (bid >> 1) & (HoN - 1);
  const int b     = bid >> 8;
  const int x0    = strip * 64;
  const int t     = threadIdx.x;
  const int px    = t & 63;
  const int grp   = t >> 6;        // 0 or 1

  // ---- stage 1: 3x3 offset conv, 18 ch x 64 px. Each wave: uniform och. ----
  {
    const int xg = x0 + px;
    for (int oi = 0; oi < 9; ++oi) {
      const int och = grp * 9 + oi;
      float acc = ob[och];
      #pragma unroll
      for (int ky = 0; ky < 3; ++ky) {
        const int iy = y + ky - 1;
        if ((unsigned)iy >= (unsigned)Hn) continue;
        #pragma unroll
        for (int kx = 0; kx < 3; ++kx) {
          const int ix = xg + kx - 1;
          if ((unsigned)ix >= (unsigned)Wn) continue;
          const float* xp = x + ((size_t)(b * Cn) * Hn + iy) * Wn + ix;
          const float* wp = ow + och * CK + ky * 3 + kx;
          #pragma unroll 8
          for (int c = 0; c < Cn; ++c)
            acc += xp[c * HW] * wp[c * K2n];
        }
      }
      s_off[och * 64 + px] = acc;
    }
  }
  __syncthreads();

  // ---- stage 2: bilinear precompute; fold OOB masks into weights,
  //      clamp indices -> 4 unconditional gather offsets per (tap,px) ----
  for (int i = t; i < K2n * 64; i += 128) {
    const int tap = i >> 6;
    const int pp  = i & 63;
    const float dy  = s_off[(tap * 2 + 0) * 64 + pp];
    const float dx  = s_off[(tap * 2 + 1) * 64 + pp];
    const float py  = dy + (float)(y - 1 + tap / 3);
    const float pxx = dx + (float)(x0 + pp - 1 + tap % 3);
    const float y0f = floorf(py);
    const float x0f = floorf(pxx);
    const float wy  = py - y0f;
    const float wx  = pxx - x0f;
    const int iy0 = (int)y0f;
    const int ix0 = (int)x0f;
    const float vy0 = ((unsigned)iy0       < (unsigned)Hn) ? 1.f : 0.f;
    const float vy1 = ((unsigned)(iy0 + 1) < (unsigned)Hn) ? 1.f : 0.f;
    const float vx0 = ((unsigned)ix0       < (unsigned)Wn) ? 1.f : 0.f;
    const float vx1 = ((unsigned)(ix0 + 1) < (unsigned)Wn) ? 1.f : 0.f;
    const int r0 = min(max(iy0, 0),     Hn - 1) * Wn;
    const int r1 = min(max(iy0 + 1, 0), Hn - 1) * Wn;
    const int c0 = min(max(ix0, 0),     Wn - 1);
    const int c1 = min(max(ix0 + 1, 0), Wn - 1);
    s_ad[i * 4 + 0] = r0 + c0;
    s_ad[i * 4 + 1] = r0 + c1;
    s_ad[i * 4 + 2] = r1 + c0;
    s_ad[i * 4 + 3] = r1 + c1;
    s_bw[i * 4 + 0] = (1.f - wy) * (1.f - wx) * vy0 * vx0;
    s_bw[i * 4 + 1] = (1.f - wy) * wx         * vy0 * vx1;
    s_bw[i * 4 + 2] = wy * (1.f - wx)         * vy1 * vx0;
    s_bw[i * 4 + 3] = wy * wx                 * vy1 * vx1;
  }
  __syncthreads();

  // ---- stage 3: per-tap sample + WMMA contraction ----
  const int lane  = t & 31;
  const int wv    = t >> 5;            // wave id -> O-chunk
  const int ln16  = lane & 15;
  const int prow  = lane >> 4;         // pair-row select for upper half-wave
  const int obase = wv * 16;

  v8f acc0 = {}, acc1 = {}, acc2 = {}, acc3 = {};

  for (int tap = 0; tap < K2n; ++tap) {
    // build sampled tile: thread owns (px, c = grp + 2j); constants in regs
    {
      const int ii = tap * 64 + px;
      const int   a0 = s_ad[ii * 4 + 0], a1 = s_ad[ii * 4 + 1];
      const int   a2 = s_ad[ii * 4 + 2], a3 = s_ad[ii * 4 + 3];
      const float w0 = s_bw[ii * 4 + 0], w1 = s_bw[ii * 4 + 1];
      const float w2 = s_bw[ii * 4 + 2], w3 = s_bw[ii * 4 + 3];
      const float* xb = x + (size_t)(b * Cn + grp) * HW;
      float* sp = &sS[px * 2 + grp];
      #pragma unroll 4
      for (int j = 0; j < 32; ++j) {
        const float* xc = xb + (size_t)(2 * j) * HW;
        sp[j * PSTR] = w0 * xc[a0] + w1 * xc[a1] + w2 * xc[a2] + w3 * xc[a3];
      }
    }
    __syncthreads();

    // 16(O) x 64(px) tile per wave; K chunks of 4 over c
    const float* pwt = pw + tap * 4096;
    #pragma unroll 4
    for (int cc = 0; cc < 16; ++cc) {
      // A fragment: one coalesced b64 per lane from packed weights
      const v2f a = *(const v2f*)(pwt + cc * 256 + wv * 64 + lane * 2);
      // B fragments: c-paired layout -> one aligned ds_load_b64 each,
      // half-waves hit disjoint bank halves (PSTR ≡ 32 mod 64)
      const float* sb = &sS[(cc * 2 + prow) * PSTR + ln16 * 2];
      const v2f b0 = *(const v2f*)(sb);
      const v2f b1 = *(const v2f*)(sb + 32);
      const v2f b2 = *(const v2f*)(sb + 64);
      const v2f b3 = *(const v2f*)(sb + 96);
      acc0 = __builtin_amdgcn_wmma_f32_16x16x4_f32(false, a, false, b0, (short)0, acc0, false, false);
      acc1 = __builtin_amdgcn_wmma_f32_16x16x4_f32(false, a, false, b1, (short)0, acc1, false, false);
      acc2 = __builtin_amdgcn_wmma_f32_16x16x4_f32(false, a, false, b2, (short)0, acc2, false, false);
      acc3 = __builtin_amdgcn_wmma_f32_16x16x4_f32(false, a, false, b3, (short)0, acc3, false, false);
    }
    __syncthreads();
  }

  // ---- stage 4: bias + store (C/D layout: VGPR i -> rows i and i+8) ----
  const int Mhi = prow * 8;
  #pragma unroll
  for (int i = 0; i < 8; ++i) {
    const int o = obase + i + Mhi;
    const float bias = db[o];
    float* orow = out + ((size_t)(b * On + o) * HoN + y) * WoN + x0 + ln16;
    orow[0]  = acc0[i] + bias;
    orow[16] = acc1[i] + bias;
    orow[32] = acc2[i] + bias;
    orow[48] = acc3[i] + bias;
  }
}

extern "C" void kernel_launch(void* const* d_in, const int* in_sizes, int n_in,
                              void* d_out, int out_size, void* d_ws, size_t ws_size,
                              hipStream_t stream) {
  const float* x  = (const float*)d_in[0];   // (4,64,128,128)
  const float* ow = (const float*)d_in[1];   // (18,64,3,3)
  const float* ob = (const float*)d_in[2];   // (18,)
  const float* dw = (const float*)d_in[3];   // (64,64,3,3)
  const float* db = (const float*)d_in[4];   // (64,)
  float* out = (float*)d_out;                // (4,64,128,128)
  float* pw  = (float*)d_ws;                 // 36864 floats = 147456 B

  repack_w<<<144, 256, 0, stream>>>(dw, pw);
  deform_fused<<<Bn * HoN * (WoN / 64), 128, 0, stream>>>(x, ow, ob, pw, db, out);
}